// BaselineCNN_48722109006500
// MI455X (gfx1250) — compile-verified
//
#include <hip/hip_runtime.h>
#include <hip/hip_bf16.h>

typedef float v2f __attribute__((ext_vector_type(2)));
typedef float v8f __attribute__((ext_vector_type(8)));

#define N_NODES 100000
#define N_EDGES 1600000
#define F_IN    10
#define HDIM    64
#define NLAYER  3
#define NGRAPH  2000
#define LN_EPS  1e-5f

// ---------------------------------------------------------------- utilities
__global__ void k_zero(float* p, int n) {
    int i = blockIdx.x * blockDim.x + threadIdx.x;
    for (; i < n; i += gridDim.x * blockDim.x) p[i] = 0.0f;
}

// ------------------------------------------------- degree / norm computation
__global__ void k_deg_scatter(const long long* __restrict__ ei, float* __restrict__ deg) {
    int e = blockIdx.x * blockDim.x + threadIdx.x;
    if (e >= N_EDGES) return;
    long long d = ei[N_EDGES + e];
    atomicAdd(&deg[d], 1.0f);
}

__global__ void k_norm(float* __restrict__ deg) {
    int n = blockIdx.x * blockDim.x + threadIdx.x;
    if (n >= N_NODES) return;
    deg[n] = rsqrtf(deg[n] + 1.0f);
}

// ----------------------------------------- embedding: LN(x @ W_emb + b_emb)
__global__ void k_embed(const float* __restrict__ x, const float* __restrict__ W,
                        const float* __restrict__ b, const float* __restrict__ lg,
                        const float* __restrict__ lb, float* __restrict__ h) {
    int n = blockIdx.x;
    int j = threadIdx.x;           // 0..63
    float acc = b[j];
#pragma unroll
    for (int k = 0; k < F_IN; ++k) acc += x[n * F_IN + k] * W[k * HDIM + j];

    __shared__ float red[HDIM];
    __shared__ float mv[2];
    red[j] = acc; __syncthreads();
    for (int s = HDIM / 2; s > 0; s >>= 1) { if (j < s) red[j] += red[j + s]; __syncthreads(); }
    if (j == 0) mv[0] = red[0] * (1.0f / HDIM);
    __syncthreads();
    float d = acc - mv[0];
    red[j] = d * d; __syncthreads();
    for (int s = HDIM / 2; s > 0; s >>= 1) { if (j < s) red[j] += red[j + s]; __syncthreads(); }
    if (j == 0) mv[1] = red[0] * (1.0f / HDIM);
    __syncthreads();
    h[(size_t)n * HDIM + j] = d * rsqrtf(mv[1] + LN_EPS) * lg[j] + lb[j];
}

// ------------------------------------- WMMA fp32 GEMM: Out[rows,C] = A[rows,64] @ B[64,C]
// One wave computes a 16-row x C strip using V_WMMA_F32_16X16X4_F32.
template <int CT>  // number of 16-wide column tiles (C = CT*16)
__global__ void k_wmma_gemm(const float* __restrict__ A, const float* __restrict__ B,
                            float* __restrict__ Out, int nrows, int K) {
    const int C = CT * 16;
    int wave   = blockIdx.x * (blockDim.x >> 5) + (threadIdx.x >> 5);
    int strips = nrows >> 4;                 // nrows is a multiple of 16
    if (wave >= strips) return;              // wave-uniform exit (EXEC all-ones below)
    int lane = threadIdx.x & 31;
    int hi   = lane >> 4;                    // 0: K lo pair / M 0..7, 1: K hi pair / M 8..15
    int lo   = lane & 15;
    int row  = (wave << 4) + lo;

    v8f acc[CT];
#pragma unroll
    for (int c = 0; c < CT; ++c)
#pragma unroll
        for (int r = 0; r < 8; ++r) acc[c][r] = 0.0f;

    for (int k0 = 0; k0 < K; k0 += 4) {
        int ka = k0 + (hi << 1);
        v2f a = *(const v2f*)(A + (size_t)row * K + ka);
#pragma unroll
        for (int c = 0; c < CT; ++c) {
            int col = (c << 4) + lo;
            v2f bv;
            bv.x = B[(size_t)ka * C + col];
            bv.y = B[(size_t)(ka + 1) * C + col];
            acc[c] = __builtin_amdgcn_wmma_f32_16x16x4_f32(
                /*neg_a=*/false, a, /*neg_b=*/false, bv,
                /*c_mod=*/(short)0, acc[c], /*reuse_a=*/false, /*reuse_b=*/false);
        }
    }

#pragma unroll
    for (int c = 0; c < CT; ++c) {
        int col = (c << 4) + lo;
#pragma unroll
        for (int r = 0; r < 8; ++r) {
            int rr = (wave << 4) + r + (hi << 3);
            Out[(size_t)rr * C + col] = acc[c][r];
        }
    }
}

// ------------------------------------------------ GCN aggregation (self loop)
__global__ void k_selfloop_init(const float* __restrict__ tmp, const float* __restrict__ nrm,
                                float* __restrict__ agg) {
    size_t i = (size_t)blockIdx.x * blockDim.x + threadIdx.x;
    if (i >= (size_t)N_NODES * HDIM) return;
    int n = (int)(i >> 6);
    float w = nrm[n];
    agg[i] = tmp[i] * w * w;
}

// one wave per edge; each lane moves 2 channels (float2 read, 2 f32 atomics)
__global__ void k_edge_scatter(const long long* __restrict__ ei, const float* __restrict__ tmp,
                               const float* __restrict__ nrm, float* __restrict__ agg) {
    int wid = (int)(((size_t)blockIdx.x * blockDim.x + threadIdx.x) >> 5);
    if (wid >= N_EDGES) return;
    int lane = threadIdx.x & 31;
    long long s = ei[wid];
    long long d = ei[N_EDGES + wid];
    float coeff = nrm[s] * nrm[d];
    const float* sp = tmp + (size_t)s * HDIM + lane * 2;
    float2 v = *(const float2*)sp;
    float* dp = agg + (size_t)d * HDIM + lane * 2;
    atomicAdd(dp + 0, v.x * coeff);
    atomicAdd(dp + 1, v.y * coeff);
}

// ----------------------------------------- BatchNorm stats over N (relu(agg+b))
__global__ void k_bn_stats(const float* __restrict__ agg, const float* __restrict__ bg,
                           float* __restrict__ stats) {
    int c   = threadIdx.x & 63;
    int grp = threadIdx.x >> 6;  // 0..3
    float s = 0.0f, q = 0.0f;
    for (int row = blockIdx.x * 4 + grp; row < N_NODES; row += gridDim.x * 4) {
        float v = agg[(size_t)row * HDIM + c] + bg[c];
        v = fmaxf(v, 0.0f);
        s += v; q += v * v;
    }
    __shared__ float S[256], Q[256];
    S[threadIdx.x] = s; Q[threadIdx.x] = q; __syncthreads();
    if (grp == 0) {
        s = S[c] + S[64 + c] + S[128 + c] + S[192 + c];
        q = Q[c] + Q[64 + c] + Q[128 + c] + Q[192 + c];
        atomicAdd(&stats[c], s);
        atomicAdd(&stats[64 + c], q);
    }
}

__global__ void k_bn_apply(const float* __restrict__ agg, const float* __restrict__ bg,
                           const float* __restrict__ stats, const float* __restrict__ bng,
                           const float* __restrict__ bnb, float* __restrict__ h) {
    size_t i = (size_t)blockIdx.x * blockDim.x + threadIdx.x;
    if (i >= (size_t)N_NODES * HDIM) return;
    int c = (int)(i & 63);
    const float invN = 1.0f / (float)N_NODES;
    float m   = stats[c] * invN;
    float var = stats[64 + c] * invN - m * m;
    float v = fmaxf(agg[i] + bg[c], 0.0f);
    h[i] += (v - m) * rsqrtf(var + LN_EPS) * bng[c] + bnb[c];
}

// ------------------------------------------------------- global mean pooling
__global__ void k_pool_scatter(const float* __restrict__ h, const long long* __restrict__ batch,
                               float* __restrict__ pool, float* __restrict__ cnt) {
    size_t i = (size_t)blockIdx.x * blockDim.x + threadIdx.x;
    if (i >= (size_t)N_NODES * HDIM) return;
    int n = (int)(i >> 6);
    int c = (int)(i & 63);
    long long g = batch[n];
    atomicAdd(&pool[(size_t)g * HDIM + c], h[i]);
    if (c == 0) atomicAdd(&cnt[g], 1.0f);
}

__global__ void k_pool_div(float* __restrict__ pool, const float* __restrict__ cnt) {
    int i = blockIdx.x * blockDim.x + threadIdx.x;
    if (i >= NGRAPH * HDIM) return;
    pool[i] /= fmaxf(cnt[i >> 6], 1.0f);
}

// --------------------------------------------- head: relu(LN(in + bias)) rows
template <int C>
__global__ void k_ln_relu(const float* __restrict__ in, const float* __restrict__ bias,
                          const float* __restrict__ lg, const float* __restrict__ lb,
                          float* __restrict__ out) {
    int n = blockIdx.x;
    int j = threadIdx.x;           // 0..C-1
    float acc = in[(size_t)n * C + j] + bias[j];
    __shared__ float red[64];
    __shared__ float mv[2];
    red[j] = acc; __syncthreads();
    for (int s = C / 2; s > 0; s >>= 1) { if (j < s) red[j] += red[j + s]; __syncthreads(); }
    if (j == 0) mv[0] = red[0] * (1.0f / C);
    __syncthreads();
    float d = acc - mv[0];
    red[j] = d * d; __syncthreads();
    for (int s = C / 2; s > 0; s >>= 1) { if (j < s) red[j] += red[j + s]; __syncthreads(); }
    if (j == 0) mv[1] = red[0] * (1.0f / C);
    __syncthreads();
    float v = d * rsqrtf(mv[1] + LN_EPS) * lg[j] + lb[j];
    out[(size_t)n * C + j] = fmaxf(v, 0.0f);
}

__global__ void k_final(const float* __restrict__ g2, const float* __restrict__ W3,
                        const float* __restrict__ b3, float* __restrict__ out) {
    int g = blockIdx.x * blockDim.x + threadIdx.x;
    if (g >= NGRAPH) return;
    float acc = b3[0];
#pragma unroll
    for (int k = 0; k < HDIM / 2; ++k) acc += g2[(size_t)g * (HDIM / 2) + k] * W3[k];
    out[g] = acc;
}

// ---------------------------------------------------------------------------
extern "C" void kernel_launch(void* const* d_in, const int* in_sizes, int n_in,
                              void* d_out, int out_size, void* d_ws, size_t ws_size,
                              hipStream_t stream) {
    (void)in_sizes; (void)n_in; (void)out_size; (void)ws_size;

    const float*     x     = (const float*)d_in[0];
    const long long* ei    = (const long long*)d_in[1];
    const long long* batch = (const long long*)d_in[2];
    const float* W_emb  = (const float*)d_in[3];
    const float* b_emb  = (const float*)d_in[4];
    const float* lne_g  = (const float*)d_in[5];
    const float* lne_b  = (const float*)d_in[6];
    const float* W_gcn  = (const float*)d_in[7];   // [3,64,64]
    const float* b_gcn  = (const float*)d_in[8];   // [3,64]
    const float* bn_g   = (const float*)d_in[9];
    const float* bn_b   = (const float*)d_in[10];
    const float* W1     = (const float*)d_in[11];
    const float* b1     = (const float*)d_in[12];
    const float* ln1_g  = (const float*)d_in[13];
    const float* ln1_b  = (const float*)d_in[14];
    const float* W2     = (const float*)d_in[15];
    const float* b2     = (const float*)d_in[16];
    const float* ln2_g  = (const float*)d_in[17];
    const float* ln2_b  = (const float*)d_in[18];
    const float* W3     = (const float*)d_in[19];
    const float* b3     = (const float*)d_in[20];
    float* out = (float*)d_out;

    // workspace layout (floats)
    float* ws    = (float*)d_ws;
    float* nrm   = ws;                                   // N
    float* h     = nrm + N_NODES;                        // N*H
    float* tmp   = h + (size_t)N_NODES * HDIM;           // N*H
    float* agg   = tmp + (size_t)N_NODES * HDIM;         // N*H
    float* stats = agg + (size_t)N_NODES * HDIM;         // 128
    // head buffers reuse tmp/agg (free after the GCN layers / pooling)
    float* pool = tmp;
    float* cnt  = tmp + NGRAPH * HDIM;
    float* g1r  = agg;
    float* g1   = g1r + NGRAPH * HDIM;
    float* g2r  = g1 + NGRAPH * HDIM;
    float* g2   = g2r + NGRAPH * (HDIM / 2);

    const size_t NH = (size_t)N_NODES * HDIM;

    // degrees -> norm
    k_zero<<<1024, 256, 0, stream>>>(nrm, N_NODES);
    k_deg_scatter<<<(N_EDGES + 255) / 256, 256, 0, stream>>>(ei, nrm);
    k_norm<<<(N_NODES + 255) / 256, 256, 0, stream>>>(nrm);

    // embedding
    k_embed<<<N_NODES, HDIM, 0, stream>>>(x, W_emb, b_emb, lne_g, lne_b, h);

    // GCN layers
    const int gcn_strips = N_NODES / 16;                     // 6250
    const int gcn_blocks = (gcn_strips + 7) / 8;             // 8 waves / block
    for (int l = 0; l < NLAYER; ++l) {
        k_wmma_gemm<4><<<gcn_blocks, 256, 0, stream>>>(h, W_gcn + (size_t)l * HDIM * HDIM,
                                                       tmp, N_NODES, HDIM);
        k_selfloop_init<<<(int)((NH + 255) / 256), 256, 0, stream>>>(tmp, nrm, agg);
        k_edge_scatter<<<N_EDGES / 8, 256, 0, stream>>>(ei, tmp, nrm, agg);
        k_zero<<<1, 128, 0, stream>>>(stats, 128);
        k_bn_stats<<<512, 256, 0, stream>>>(agg, b_gcn + l * HDIM, stats);
        k_bn_apply<<<(int)((NH + 255) / 256), 256, 0, stream>>>(agg, b_gcn + l * HDIM, stats,
                                                                bn_g + l * HDIM, bn_b + l * HDIM, h);
    }

    // global mean pool
    k_zero<<<256, 256, 0, stream>>>(pool, NGRAPH * HDIM);
    k_zero<<<8, 256, 0, stream>>>(cnt, NGRAPH);
    k_pool_scatter<<<(int)((NH + 255) / 256), 256, 0, stream>>>(h, batch, pool, cnt);
    k_pool_div<<<(NGRAPH * HDIM + 255) / 256, 256, 0, stream>>>(pool, cnt);

    // head
    const int head_blocks = (NGRAPH / 16 + 7) / 8;           // 125 strips
    k_wmma_gemm<4><<<head_blocks, 256, 0, stream>>>(pool, W1, g1r, NGRAPH, HDIM);
    k_ln_relu<HDIM><<<NGRAPH, HDIM, 0, stream>>>(g1r, b1, ln1_g, ln1_b, g1);
    k_wmma_gemm<2><<<head_blocks, 256, 0, stream>>>(g1, W2, g2r, NGRAPH, HDIM);
    k_ln_relu<HDIM / 2><<<NGRAPH, HDIM / 2, 0, stream>>>(g2r, b2, ln2_g, ln2_b, g2);
    k_final<<<(NGRAPH + 255) / 256, 256, 0, stream>>>(g2, W3, b3, out);
}